// RESA_80625126080575
// MI455X (gfx1250) — compile-verified
//
#include <hip/hip_runtime.h>
#include <hip/hip_bf16.h>

#define NCH   256
#define HH    64
#define WW    160
#define HW    (HH*WW)          // 10240
#define NBATCH 8

typedef __attribute__((ext_vector_type(2))) float v2f;
typedef __attribute__((ext_vector_type(8))) float v8f;

// -------------------------------------------------------------------------
// Kernel 1: y = relu(bn(conv1x1(x)))  as a 256x256 x 256x81920 GEMM
// using V_WMMA_F32_16X16X4_F32 (fp32 to match reference numerics; stage is
// bandwidth-bound so there is nothing to gain from quantizing).
// Grid: 640 n-groups (128 px) x 16 m-tiles, n-major for L2 reuse of x
// (x = 84 MB < 192 MB L2, shared by the 16 m-tile blocks per n-group).
// Block: 256 threads = 8 waves, each wave owns one 16x16 output tile.
// -------------------------------------------------------------------------
__global__ __launch_bounds__(256) void RESA_conv1x1_bn_relu(
    const float* __restrict__ x,      // [B, C, H, W]
    const float* __restrict__ w,      // [C_out, C_in] row-major
    const float* __restrict__ bias,   // [C]
    const float* __restrict__ gamma,
    const float* __restrict__ beta,
    const float* __restrict__ mean,
    const float* __restrict__ var,
    float* __restrict__ y)            // [B, C, H, W]  (to d_ws)
{
    __shared__ float Alds[16 * 260];          // padded stride -> bank-clean b64

    const int mtile = blockIdx.x & 15;        // 16 m-tiles of 16 channels
    const int ngrp  = blockIdx.x >> 4;        // 640 groups of 128 pixels
    const int tid   = threadIdx.x;
    const int wave  = tid >> 5;
    const int lane  = tid & 31;
    const int mbase = mtile * 16;

    for (int idx = tid; idx < 16 * 256; idx += 256) {
        const int m = idx >> 8;
        const int k = idx & 255;
        Alds[m * 260 + k] = w[(mbase + m) * NCH + k];
    }
    __syncthreads();

    const int b    = ngrp / 80;               // 80 n-groups per batch image
    const int hw0  = (ngrp % 80) * 128 + wave * 16;
    const int n    = lane & 15;
    const int half = lane >> 4;               // 0: K={0,1}, 1: K={2,3}
    const int m    = lane & 15;
    const int koff = half * 2;

    const float* xb = x + (size_t)b * NCH * HW + hw0 + n;

    v8f acc = {};
#pragma unroll 4
    for (int k0 = 0; k0 < 256; k0 += 4) {
        v2f a = *(const v2f*)&Alds[m * 260 + k0 + koff];
        v2f bb;
        bb.x = xb[(size_t)(k0 + koff)     * HW];
        bb.y = xb[(size_t)(k0 + koff + 1) * HW];
        acc = __builtin_amdgcn_wmma_f32_16x16x4_f32(
            false, a, false, bb, (short)0, acc, false, false);
    }

    float* yb = y + (size_t)b * NCH * HW + hw0 + n;
#pragma unroll
    for (int r = 0; r < 8; ++r) {
        const int o = mbase + r + half * 8;
        const float s = gamma[o] * rsqrtf(var[o] + 1e-5f);
        const float t = (bias[o] - mean[o]) * s + beta[o];
        yb[(size_t)o * HW] = fmaxf(fmaf(acc[r], s, t), 0.0f);
    }
}

// -------------------------------------------------------------------------
// Kernel 2a: main refine pass (interior-exact, border fixed by kernel 2b).
// Identity: for pixels whose 3x3 window stays inside the image,
//   dw3x3(roll(y,s))[p] == zc[(p - s) mod (H,W)]
// where zc is the *circular* 3x3 conv of y. So precompute zc once per tile
// (9 FMA/entry, ~1.8 entries/pixel amortized), then each shift costs
// 1 ds_load + fma + max + fma.  ~4.5x fewer VALU/DS ops than the direct
// 12x9 masked form, and the hot loop has no masks / no divergence.
// -------------------------------------------------------------------------
__global__ __launch_bounds__(256) void RESA_refine_main(
    const float* __restrict__ x,
    const float* __restrict__ y,      // from kernel 1 (d_ws)
    const float* __restrict__ rw,     // [C,1,3,3]
    const float* __restrict__ rb,     // [C]
    const float* __restrict__ g,
    const float* __restrict__ be,
    const float* __restrict__ mu,
    const float* __restrict__ va,
    float* __restrict__ out)
{
    __shared__ float tile[16 * 168];  // y halo'd +/-4, wrapped mod H / mod W
    __shared__ float zcb[14 * 168];   // circular conv, rows rowbase-3..+10,
                                      // cols -3..162 (index cc = col+3)

    const int rt      = blockIdx.x & 7;
    const int bc      = blockIdx.x >> 3;      // b*C + c
    const int c       = bc & (NCH - 1);
    const int rowbase = rt * 8;
    const int tid     = threadIdx.x;

    // tile[r][col] = y[(rowbase-4+r) mod H][(col-4) mod W] : the +/-1
    // neighbors inside the tile are exactly the circular neighbors, so a
    // plain 3x3 conv on the tile IS the circular conv.
    const float* ybc = y + (size_t)bc * HW;
    for (int idx = tid; idx < 16 * 168; idx += 256) {
        const int r   = idx / 168;
        const int col = idx - r * 168;
        const int gr  = (rowbase - 4 + r + HH) % HH;
        const int gc  = (col - 4 + WW) % WW;
        tile[idx] = ybc[gr * WW + gc];
    }

    // Per-channel constants: c is blockIdx-derived -> scalar loads.
    float w9[9];
#pragma unroll
    for (int t = 0; t < 9; ++t) w9[t] = rw[c * 9 + t];
    const float s  = g[c] * rsqrtf(va[c] + 1e-5f);
    const float t2 = (rb[c] - mu[c]) * s + be[c];   // bn(z + rb) = z*s + t2

    const float wt1 = expf(-1.0f / 4.5f);
    const float wt2 = expf(-4.0f / 4.5f);
    const float wt3 = expf(-9.0f / 4.5f);
    const float outscale = 0.5f / (4.0f * (wt1 + wt2 + wt3));
    const float WT[12] = {wt1,wt2,wt3, wt1,wt2,wt3, wt1,wt2,wt3, wt1,wt2,wt3};
    const int   SH[12] = { 1, 2, 3, -1,-2,-3,  0, 0, 0,  0, 0, 0};
    const int   SW[12] = { 0, 0, 0,  0, 0, 0, -1,-2,-3,  1, 2, 3};

    __syncthreads();

    // Circular-conv precompute: zcb[r][cc] = zc[rowbase-3+r][cc-3]
    // consecutive tids -> consecutive cc -> conflict-free LDS.
    for (int idx = tid; idx < 14 * 166; idx += 256) {
        const int r  = idx / 166;
        const int cc = idx - r * 166;
        float z = 0.0f;
#pragma unroll
        for (int dy = 0; dy < 3; ++dy)
#pragma unroll
            for (int dx = 0; dx < 3; ++dx)
                z = fmaf(w9[dy * 3 + dx], tile[(r + dy) * 168 + cc + dx], z);
        zcb[r * 168 + cc] = z;
    }
    __syncthreads();

    for (int k = 0; k < 5; ++k) {             // 1280 px / 256 threads
        const int px   = tid + k * 256;
        const int lr   = px / 160;
        const int wcol = px - lr * 160;
        const int h    = rowbase + lr;

        float acc = 0.0f;
#pragma unroll
        for (int si = 0; si < 12; ++si) {
            // zc[(h - sh)][(w - sw)] with circular wrap baked into zcb
            const float z = zcb[(lr + 3 - SH[si]) * 168 + (wcol + 3 - SW[si])];
            acc = fmaf(WT[si], fmaxf(fmaf(z, s, t2), 0.0f), acc);
        }

        const size_t gidx = (size_t)bc * HW + (size_t)h * WW + wcol;
        out[gidx] = x[gidx] + outscale * acc;   // border pixels overwritten by 2b
    }
}

// -------------------------------------------------------------------------
// Kernel 2b: exact recompute of the 444 border pixels per (b,c) plane
// (h in {0,H-1} or w in {0,W-1}), where SAME zero-padding breaks the
// circular-conv identity. ~0.9M pixels total, y is L2-resident.
// -------------------------------------------------------------------------
__global__ __launch_bounds__(256) void RESA_refine_border(
    const float* __restrict__ x,
    const float* __restrict__ y,
    const float* __restrict__ rw,
    const float* __restrict__ rb,
    const float* __restrict__ g,
    const float* __restrict__ be,
    const float* __restrict__ mu,
    const float* __restrict__ va,
    float* __restrict__ out)
{
    const int bc = blockIdx.x;                // b*C + c, 2048 planes
    const int c  = bc & (NCH - 1);

    float w9[9];
#pragma unroll
    for (int t = 0; t < 9; ++t) w9[t] = rw[c * 9 + t];
    const float s  = g[c] * rsqrtf(va[c] + 1e-5f);
    const float t2 = (rb[c] - mu[c]) * s + be[c];

    const float wt1 = expf(-1.0f / 4.5f);
    const float wt2 = expf(-4.0f / 4.5f);
    const float wt3 = expf(-9.0f / 4.5f);
    const float outscale = 0.5f / (4.0f * (wt1 + wt2 + wt3));
    const float WT[12] = {wt1,wt2,wt3, wt1,wt2,wt3, wt1,wt2,wt3, wt1,wt2,wt3};
    const int   SH[12] = { 1, 2, 3, -1,-2,-3,  0, 0, 0,  0, 0, 0};
    const int   SW[12] = { 0, 0, 0,  0, 0, 0, -1,-2,-3,  1, 2, 3};

    const float* yp = y + (size_t)bc * HW;

    // 2*W + 2*(H-2) = 444 border pixels per plane
    for (int e = threadIdx.x; e < 444; e += 256) {
        int h, w;
        if      (e < 160) { h = 0;      w = e;       }
        else if (e < 320) { h = HH - 1; w = e - 160; }
        else if (e < 382) { w = 0;      h = e - 319; }   // h = 1..62
        else              { w = WW - 1; h = e - 381; }   // h = 1..62

        float acc = 0.0f;
#pragma unroll
        for (int si = 0; si < 12; ++si) {
            float conv = 0.0f;
#pragma unroll
            for (int dy = 0; dy < 3; ++dy) {
#pragma unroll
                for (int dx = 0; dx < 3; ++dx) {
                    const int i = h + dy - 1;            // rolled-image coord
                    const int j = w + dx - 1;
                    const bool ok = ((unsigned)i < (unsigned)HH) &&
                                    ((unsigned)j < (unsigned)WW);
                    const int ii = (i - SH[si] + HH) % HH;
                    const int jj = (j - SW[si] + WW) % WW;
                    const float v = ok ? yp[ii * WW + jj] : 0.0f;
                    conv = fmaf(w9[dy * 3 + dx], v, conv);
                }
            }
            acc = fmaf(WT[si], fmaxf(fmaf(conv, s, t2), 0.0f), acc);
        }

        const size_t gidx = (size_t)bc * HW + (size_t)h * WW + w;
        out[gidx] = x[gidx] + outscale * acc;
    }
}

extern "C" void kernel_launch(void* const* d_in, const int* in_sizes, int n_in,
                              void* d_out, int out_size, void* d_ws, size_t ws_size,
                              hipStream_t stream) {
    const float* x       = (const float*)d_in[0];
    const float* conv_w  = (const float*)d_in[1];
    const float* conv_b  = (const float*)d_in[2];
    const float* bn_g    = (const float*)d_in[3];
    const float* bn_b    = (const float*)d_in[4];
    const float* bn_m    = (const float*)d_in[5];
    const float* bn_v    = (const float*)d_in[6];
    const float* ref_w   = (const float*)d_in[7];
    const float* ref_b   = (const float*)d_in[8];
    const float* rbn_g   = (const float*)d_in[9];
    const float* rbn_b   = (const float*)d_in[10];
    const float* rbn_m   = (const float*)d_in[11];
    const float* rbn_v   = (const float*)d_in[12];
    float* outp = (float*)d_out;
    float* yws  = (float*)d_ws;   // 8*256*64*160*4 = 83.9 MB intermediate

    RESA_conv1x1_bn_relu<<<640 * 16, 256, 0, stream>>>(
        x, conv_w, conv_b, bn_g, bn_b, bn_m, bn_v, yws);

    RESA_refine_main<<<NBATCH * NCH * (HH / 8), 256, 0, stream>>>(
        x, yws, ref_w, ref_b, rbn_g, rbn_b, rbn_m, rbn_v, outp);

    RESA_refine_border<<<NBATCH * NCH, 256, 0, stream>>>(
        x, yws, ref_w, ref_b, rbn_g, rbn_b, rbn_m, rbn_v, outp);
}